// ResidualBlock_20375324852254
// MI455X (gfx1250) — compile-verified
//
#include <hip/hip_runtime.h>

#define D 128
#define BN_EPS 1e-5f

typedef __attribute__((ext_vector_type(2))) float v2f;
typedef __attribute__((ext_vector_type(8))) float v8f;

// ---------------------------------------------------------------------------
// Degree computation: deg initialized to 1.0 (self loop), +1 per incoming edge,
// then converted in place to rsqrt(deg).
// ---------------------------------------------------------------------------
__global__ void deg_init_kernel(float* __restrict__ deg, int n) {
    int i = blockIdx.x * blockDim.x + threadIdx.x;
    if (i < n) deg[i] = 1.0f;
}

__global__ void deg_count_kernel(const int* __restrict__ dst, float* __restrict__ deg, int e) {
    int i = blockIdx.x * blockDim.x + threadIdx.x;
    if (i < e) atomicAdd(&deg[dst[i]], 1.0f);
}

__global__ void dinv_kernel(float* __restrict__ deg, int n) {
    int i = blockIdx.x * blockDim.x + threadIdx.x;
    if (i < n) deg[i] = rsqrtf(deg[i]);
}

// ---------------------------------------------------------------------------
// h = x @ W^T via V_WMMA_F32_16X16X4_F32.
// One block = 256 threads = 8 waves; block covers a 16-row band of x, wave w
// produces output columns [16w, 16w+16). W (64KB) is staged in LDS transposed
// (WT[k][n]) so the B-fragment read strides lanes over n -> conflict-free.
//
// A frag (16x4 f32): lane = (half)*16 + m ; v0 = A[m][k0+2*half], v1 = A[m][k0+2*half+1]
// B frag (4x16 f32): v0 = B[k0+2*half][n],  v1 = B[k0+2*half+1][n], n = lane&15
// C/D  (16x16 f32): vgpr r : lanes 0-15 -> M=r, N=lane ; lanes 16-31 -> M=r+8
// ---------------------------------------------------------------------------
__global__ __launch_bounds__(256) void gemm_wmma_kernel(const float* __restrict__ x,
                                                        const float* __restrict__ W,
                                                        float* __restrict__ h, int n) {
    __shared__ float WT[D * D];  // 64 KB, transposed: WT[k*D + nn] = W[nn*D + k]

    for (int i = threadIdx.x; i < D * D; i += blockDim.x) {
        int r = i / D, c = i % D;          // W[r][c]
        WT[c * D + r] = W[i];
    }
    __syncthreads();

    const int row0 = blockIdx.x * 16;
    if (row0 + 16 > n) return;             // full tiles only (tail kernel handles rest)

    const int lane = threadIdx.x & 31;
    const int wave = threadIdx.x >> 5;     // 0..7 -> column tile
    const int n0   = wave * 16;
    const int mr   = lane & 15;            // m (for A) / n (for B) within tile
    const int koff = (lane >> 4) * 2;      // K sub-offset per lane half

    const float* xrow = x + (size_t)(row0 + mr) * D;

    v8f acc = {};
    for (int k0 = 0; k0 < D; k0 += 4) {
        v2f a, b;
        const float2 av = *(const float2*)(xrow + k0 + koff);   // 8B aligned
        a.x = av.x;  a.y = av.y;
        b.x = WT[(k0 + koff) * D + n0 + mr];
        b.y = WT[(k0 + koff + 1) * D + n0 + mr];
        acc = __builtin_amdgcn_wmma_f32_16x16x4_f32(
            /*neg_a=*/false, a, /*neg_b=*/false, b,
            /*c_mod=*/(short)0, acc, /*reuse_a=*/false, /*reuse_b=*/false);
    }

    const int mbase = row0 + (lane >> 4) * 8;
    float* hp = h + (size_t)mbase * D + n0 + mr;
#pragma unroll
    for (int i = 0; i < 8; ++i) hp[(size_t)i * D] = acc[i];
}

// Scalar tail for rows not covered by full 16-row tiles (0 rows when n%16==0).
__global__ void gemm_tail_kernel(const float* __restrict__ x, const float* __restrict__ W,
                                 float* __restrict__ h, int row_start, int n) {
    int idx = blockIdx.x * blockDim.x + threadIdx.x;   // over (n-row_start)*D
    int row = row_start + idx / D;
    int col = idx % D;
    if (row >= n) return;
    const float* xr = x + (size_t)row * D;
    const float* wr = W + (size_t)col * D;
    float s = 0.f;
    for (int k = 0; k < D; ++k) s += xr[k] * wr[k];
    h[(size_t)row * D + col] = s;
}

// ---------------------------------------------------------------------------
// agg[i][:] = h[i][:] * dinv[i]^2 + b[:]   (self-loop contribution + bias)
// ---------------------------------------------------------------------------
__global__ void agg_init_kernel(const float* __restrict__ h, const float* __restrict__ dinv,
                                const float* __restrict__ b, float* __restrict__ agg, int n) {
    int idx = blockIdx.x * blockDim.x + threadIdx.x;   // over n * (D/4)
    int total = n * (D / 4);
    if (idx >= total) return;
    int row  = idx / (D / 4);
    int col4 = (idx % (D / 4)) * 4;
    float di   = dinv[row];
    float norm = di * di;
    float4 hv = *(const float4*)(h + (size_t)row * D + col4);
    float4 bv = *(const float4*)(b + col4);
    float4 o;
    o.x = hv.x * norm + bv.x;
    o.y = hv.y * norm + bv.y;
    o.z = hv.z * norm + bv.z;
    o.w = hv.w * norm + bv.w;
    *(float4*)(agg + (size_t)row * D + col4) = o;
}

// ---------------------------------------------------------------------------
// Edge scatter: one wave32 per edge; lane owns 4 consecutive floats of the row.
// agg[dst] += h[src] * (dinv[src]*dinv[dst])  via global f32 atomics (L2-resident).
// ---------------------------------------------------------------------------
__global__ __launch_bounds__(256) void scatter_kernel(const int* __restrict__ src,
                                                      const int* __restrict__ dst,
                                                      const float* __restrict__ h,
                                                      const float* __restrict__ dinv,
                                                      float* __restrict__ agg, int e) {
    int wid  = (blockIdx.x * blockDim.x + threadIdx.x) >> 5;
    int lane = threadIdx.x & 31;
    if (wid >= e) return;
    int s = src[wid];
    int d = dst[wid];
    float norm = dinv[s] * dinv[d];
    float4 v = *(const float4*)(h + (size_t)s * D + lane * 4);
    float* ap = agg + (size_t)d * D + lane * 4;
    atomicAdd(ap + 0, v.x * norm);
    atomicAdd(ap + 1, v.y * norm);
    atomicAdd(ap + 2, v.z * norm);
    atomicAdd(ap + 3, v.w * norm);
}

// ---------------------------------------------------------------------------
// out = relu(relu(gamma*(agg-mean)*rsqrt(var+eps)+beta) + x), in place on agg.
// ---------------------------------------------------------------------------
__global__ void bn_relu_kernel(const float* __restrict__ x, const float* __restrict__ gamma,
                               const float* __restrict__ beta, const float* __restrict__ mean,
                               const float* __restrict__ var, float* __restrict__ agg, int n) {
    int idx = blockIdx.x * blockDim.x + threadIdx.x;   // over n * (D/4)
    int total = n * (D / 4);
    if (idx >= total) return;
    int row  = idx / (D / 4);
    int col4 = (idx % (D / 4)) * 4;
    float4 a = *(const float4*)(agg + (size_t)row * D + col4);
    float4 xv = *(const float4*)(x + (size_t)row * D + col4);
    float4 g = *(const float4*)(gamma + col4);
    float4 be = *(const float4*)(beta + col4);
    float4 m = *(const float4*)(mean + col4);
    float4 vv = *(const float4*)(var + col4);
    float4 o;
    o.x = fmaxf(fmaxf(g.x * (a.x - m.x) * rsqrtf(vv.x + BN_EPS) + be.x, 0.f) + xv.x, 0.f);
    o.y = fmaxf(fmaxf(g.y * (a.y - m.y) * rsqrtf(vv.y + BN_EPS) + be.y, 0.f) + xv.y, 0.f);
    o.z = fmaxf(fmaxf(g.z * (a.z - m.z) * rsqrtf(vv.z + BN_EPS) + be.z, 0.f) + xv.z, 0.f);
    o.w = fmaxf(fmaxf(g.w * (a.w - m.w) * rsqrtf(vv.w + BN_EPS) + be.w, 0.f) + xv.w, 0.f);
    *(float4*)(agg + (size_t)row * D + col4) = o;
}

// ---------------------------------------------------------------------------
extern "C" void kernel_launch(void* const* d_in, const int* in_sizes, int n_in,
                              void* d_out, int out_size, void* d_ws, size_t ws_size,
                              hipStream_t stream) {
    const float* x        = (const float*)d_in[0];
    const int*   ei       = (const int*)d_in[1];
    const float* W        = (const float*)d_in[2];
    const float* b        = (const float*)d_in[3];
    const float* gamma    = (const float*)d_in[4];
    const float* beta     = (const float*)d_in[5];
    const float* run_mean = (const float*)d_in[6];
    const float* run_var  = (const float*)d_in[7];

    const int n = in_sizes[0] / D;
    const int e = in_sizes[1] / 2;
    const int* src = ei;        // edge_index[0]
    const int* dst = ei + e;    // edge_index[1]

    float* h    = (float*)d_ws;                 // n*D floats
    float* dinv = h + (size_t)n * D;            // n floats
    float* agg  = (float*)d_out;                // n*D floats (accumulator, finalized in place)

    // 1) degrees -> dinv
    deg_init_kernel<<<(n + 255) / 256, 256, 0, stream>>>(dinv, n);
    deg_count_kernel<<<(e + 255) / 256, 256, 0, stream>>>(dst, dinv, e);
    dinv_kernel<<<(n + 255) / 256, 256, 0, stream>>>(dinv, n);

    // 2) h = x @ W^T  (WMMA f32 16x16x4)
    int full_tiles = n / 16;
    if (full_tiles > 0)
        gemm_wmma_kernel<<<full_tiles, 256, 0, stream>>>(x, W, h, n);
    int row_start = full_tiles * 16;
    int tail_rows = n - row_start;
    if (tail_rows > 0)
        gemm_tail_kernel<<<(tail_rows * D + 255) / 256, 256, 0, stream>>>(x, W, h, row_start, n);

    // 3) agg = self-loop + bias
    agg_init_kernel<<<((n * (D / 4)) + 255) / 256, 256, 0, stream>>>(h, dinv, b, agg, n);

    // 4) edge scatter-add (one wave per edge, 8 edges per 256-thread block)
    scatter_kernel<<<(e + 7) / 8, 256, 0, stream>>>(src, dst, h, dinv, agg, e);

    // 5) BN(eval) + ReLU + residual + ReLU, in place
    bn_relu_kernel<<<((n * (D / 4)) + 255) / 256, 256, 0, stream>>>(
        x, gamma, beta, run_mean, run_var, agg, n);
}